// Decoder_31791347925413
// MI455X (gfx1250) — compile-verified
//
#include <hip/hip_runtime.h>
#include <math.h>

typedef __attribute__((ext_vector_type(2))) float v2f;
typedef __attribute__((ext_vector_type(8))) float v8f;

#define NIMG 128
#define LVEC 15
#define SIZE 64
#define PIX  (SIZE*SIZE)        /* 4096 */
#define SKW  256
#define SKTEX (SKW*SKW)         /* 65536 */
#define NK   33
#define KPAD 36
#define PARAM_STRIDE 128
#define IBATCH 4                /* images per WMMA register block */

/* output offsets (flat, return order: rgb, shade, spec, blood, mel, b, raw) */
#define RGB_OFF   0
#define SHADE_OFF (NIMG*3*PIX)
#define SPEC_OFF  (SHADE_OFF + NIMG*PIX)
#define BLOOD_OFF (SPEC_OFF + NIMG*3*PIX)
#define MEL_OFF   (BLOOD_OFF + NIMG*PIX)
#define B_OFF     (MEL_OFF + NIMG*PIX)
#define RAW_OFF   (B_OFF + NIMG*2)

__device__ __constant__ float c_S0[NK] = {
  94.8f,104.8f,105.9f,96.8f,113.9f,125.6f,125.5f,121.3f,121.3f,113.5f,113.1f,
  110.8f,106.5f,108.8f,105.3f,104.4f,100.0f,96.0f,95.1f,89.1f,90.5f,90.3f,
  88.4f,84.0f,85.1f,81.9f,82.6f,84.9f,81.3f,71.9f,74.3f,76.4f,63.3f};
__device__ __constant__ float c_S1[NK] = {
  43.4f,46.3f,43.9f,37.1f,36.7f,35.9f,32.6f,27.9f,24.3f,20.1f,16.2f,13.2f,
  8.6f,6.1f,4.2f,1.9f,0.0f,-1.6f,-3.5f,-3.5f,-5.8f,-7.2f,-8.6f,-9.5f,
  -10.9f,-10.7f,-12.0f,-14.0f,-13.6f,-12.0f,-13.3f,-12.9f,-10.6f};
__device__ __constant__ float c_S2[NK] = {
  -1.1f,-0.5f,-0.7f,-1.2f,-2.6f,-2.9f,-2.8f,-2.6f,-2.6f,-1.8f,-1.5f,-1.3f,
  -1.2f,-1.0f,-0.5f,-0.3f,0.0f,0.2f,0.5f,2.1f,3.2f,4.1f,4.7f,5.1f,6.7f,
  7.3f,8.6f,9.8f,10.2f,8.3f,9.6f,8.5f,7.0f};
__device__ __constant__ float c_M[9] = {
  3.2406f,-1.537f,-0.498f, -0.968f,1.8758f,0.0415f, 0.0557f,-0.204f,1.057f};

/* ---------------- Kernel 1: per-image constants ---------------- */
__global__ void setup_kernel(const float* __restrict__ lighting,
                             const float* __restrict__ illA,
                             const float* __restrict__ illF,
                             const float* __restrict__ pcaMeans,
                             const float* __restrict__ pcaComp,
                             const float* __restrict__ tmatrix,
                             float* __restrict__ params,
                             float* __restrict__ out)
{
  int n = blockIdx.x * blockDim.x + threadIdx.x;
  if (n >= NIMG) return;
  const float* L = lighting + n*(LVEC+2);

  /* softmax over lp[0:14] */
  float lp[14]; float mx = -1e30f;
  for (int i=0;i<14;i++){ lp[i]=L[i]; mx=fmaxf(mx,lp[i]); }
  float ssum=0.f;
  for (int i=0;i<14;i++){ lp[i]=expf(lp[i]-mx); ssum+=lp[i]; }
  float sinv = 1.0f/ssum;
  float wA = lp[0]*sinv, wD = lp[1]*sinv;
  float fW[12];
  for (int j=0;j<12;j++) fW[j]=lp[j+2]*sinv;

  float ct = 1.0f/(1.0f+expf(-L[14]));
  float b0 = 6.0f/(1.0f+expf(-L[15])) - 3.0f;
  float b1 = 6.0f/(1.0f+expf(-L[16])) - 3.0f;
  out[B_OFF + n*2 + 0] = b0;
  out[B_OFF + n*2 + 1] = b1;

  /* illuminance_d */
  float t  = ct*21000.0f + 4000.0f;
  float it = 1.0f/t;
  float it2 = it*it, it3 = it2*it;
  float x1 = -4.607e9f*it3 + 2.9678e6f*it2 + 99.11f*it + 0.244063f;
  float x2 = -2.0064e9f*it3 + 1.9018e6f*it2 + 247.48f*it + 0.23704f;
  float x  = (t <= 7000.0f) ? x1 : x2;
  float y  = -3.0f*x*x + 2.87f*x - 0.275f;
  float m  = 0.0241f + 0.2562f*x - 0.7341f*y;
  float m1 = (-1.3515f - 1.7703f*x + 5.9114f*y)/m;
  float m2 = (0.03f   - 31.4424f*x + 30.0717f*y)/m;
  float dcurve[NK]; float dsum = 0.f;
  for (int k=0;k<NK;k++){ float s = c_S0[k] + m1*c_S1[k] + m2*c_S2[k]; dcurve[k]=s; dsum+=s; }
  float dinv = 1.0f/dsum;

  /* e (normalized) */
  float e[NK]; float esum = 0.f;
  for (int k=0;k<NK;k++){
    float v = wA*illA[k] + wD*dcurve[k]*dinv;
    for (int j=0;j<12;j++) v += fW[j]*illF[k*12+j];
    e[k]=v; esum+=v;
  }
  float einv = 1.0f/esum;

  /* S = relu(b @ pcaComp^T + means); A = S*e (padded to KPAD); lightColor */
  float* prm = params + (size_t)n*PARAM_STRIDE;
  float lc[3];
  for (int c=0;c<3;c++){
    float acc = 0.f;
    for (int k=0;k<NK;k++){
      int i = c*NK + k;
      float s = b0*pcaComp[i*2+0] + b1*pcaComp[i*2+1] + pcaMeans[i];
      s = fmaxf(s, 0.0f);
      acc += s;
      prm[c*KPAD + k] = s * (e[k]*einv);
    }
    for (int k=NK;k<KPAD;k++) prm[c*KPAD + k] = 0.0f;
    lc[c] = acc;
  }
  for (int c=0;c<3;c++){ prm[108+c] = lc[c]; prm[111+c] = 1.0f/lc[c]; }

  /* T = bilerp(tmatrix) at (b/3); Q = XYZ2RGB * T^T folded */
  float gx = (b0/3.0f + 1.0f)*0.5f*127.0f;
  float gy = (b1/3.0f + 1.0f)*0.5f*127.0f;
  float fx = floorf(gx), fy = floorf(gy);
  float wx = gx - fx,    wy = gy - fy;
  int X0 = min(max((int)fx,0),127), Y0 = min(max((int)fy,0),127);
  int X1 = min(X0+1,127),           Y1 = min(Y0+1,127);
  float w00=(1.f-wx)*(1.f-wy), w01=wx*(1.f-wy), w10=(1.f-wx)*wy, w11=wx*wy;
  float Tm[9];
  for (int c=0;c<9;c++){
    const float* pl = tmatrix + (size_t)c*128*128;
    Tm[c] = pl[Y0*128+X0]*w00 + pl[Y0*128+X1]*w01 + pl[Y1*128+X0]*w10 + pl[Y1*128+X1]*w11;
  }
  for (int o=0;o<3;o++)
    for (int c=0;c<3;c++)
      prm[114 + o*3 + c] = c_M[o*3+0]*Tm[3*c+0] + c_M[o*3+1]*Tm[3*c+1] + c_M[o*3+2]*Tm[3*c+2];
}

/* ------- Kernel 2: P[c][tex] = sum_k A[c][k]*skin[k][tex] via WMMA -------
 * M=16 (rows 0..2 live), K=32 via 8 WMMA k-steps, K=32 tail via 12 scalar
 * FMAs (rank-1 update of 3 live rows; ak32 coefs come from uniform loads ->
 * SGPRs). Register-blocked over IBATCH=4 images: each B fragment (shared
 * skinColor) feeds 4 independent accumulator chains -> 4x less skin L2
 * traffic, 4 independent WMMAs per k-step, no conditionals in the hot loop.
 * f32 A 16x4 layout: VGPR0 = K {0 | 2} across lane halves, VGPR1 = K {1 | 3}.
 * f32 D 16x16 layout: VGPR j -> row M=j (lanes 0-15), M=j+8 (lanes 16-31). */
__global__ void gemm_kernel(const float* __restrict__ params,
                            const float* __restrict__ skin,
                            float* __restrict__ P, int imgStart, int imgCount)
{
  int g     = blockIdx.y;                    /* image group (IBATCH images) */
  int tg    = blockIdx.x;                    /* 0..15 tile groups */
  int wave  = threadIdx.x >> 5;              /* 0..7 */
  int lane  = threadIdx.x & 31;
  int mrow  = lane & 15;
  int khalf = (lane >> 4) ? 2 : 0;

  v2f   a[IBATCH][8];
  float ak32[IBATCH][3];                     /* uniform -> SGPRs */
  bool  valid[IBATCH];
  #pragma unroll
  for (int i=0;i<IBATCH;i++){
    int li = g*IBATCH + i;                   /* chunk-local image */
    valid[i] = (li < imgCount);
    const float* A = params + (size_t)(imgStart + li)*PARAM_STRIDE;
    #pragma unroll
    for (int kk=0; kk<8; kk++){
      int k = kk*4 + khalf;
      float ax = 0.0f, ay = 0.0f;
      if (valid[i] && mrow < 3){ ax = A[mrow*KPAD + k]; ay = A[mrow*KPAD + k + 1]; }
      a[i][kk].x = ax; a[i][kk].y = ay;
    }
    #pragma unroll
    for (int c=0;c<3;c++)
      ak32[i][c] = valid[i] ? A[c*KPAD + 32] : 0.0f;   /* uniform address */
  }

  for (int t = wave; t < 256; t += 8){
    int base = (tg*256 + t)*16;
    v8f acc[IBATCH];
    #pragma unroll
    for (int i=0;i<IBATCH;i++) acc[i] = (v8f){0.f,0.f,0.f,0.f,0.f,0.f,0.f,0.f};

    float b32v = skin[(size_t)32*SKTEX + base + mrow];  /* K=32 column */

    #pragma unroll
    for (int kk=0; kk<8; kk++){
      int k = kk*4 + khalf;
      v2f bf;
      bf.x = skin[(size_t)k*SKTEX     + base + mrow];
      bf.y = skin[(size_t)(k+1)*SKTEX + base + mrow];
      #pragma unroll
      for (int i=0;i<IBATCH;i++)
        acc[i] = __builtin_amdgcn_wmma_f32_16x16x4_f32(false, a[i][kk], false, bf,
                                                       (short)0, acc[i], false, false);
    }

    /* K=32 rank-1 tail on the 3 live rows (lanes 16-31 touch dead rows) */
    #pragma unroll
    for (int i=0;i<IBATCH;i++){
      acc[i][0] += ak32[i][0] * b32v;
      acc[i][1] += ak32[i][1] * b32v;
      acc[i][2] += ak32[i][2] * b32v;
    }

    if (lane < 16){
      #pragma unroll
      for (int i=0;i<IBATCH;i++){
        if (valid[i]){
          float* Pimg = P + (size_t)(g*IBATCH + i)*3*SKTEX;
          Pimg[0*SKTEX + base + lane] = acc[i][0];  /* M=0 : channel 0 */
          Pimg[1*SKTEX + base + lane] = acc[i][1];  /* M=1 : channel 1 */
          Pimg[2*SKTEX + base + lane] = acc[i][2];  /* M=2 : channel 2 */
        }
      }
    }
  }
}

/* ---------------- Kernel 3: per-pixel pipeline ---------------- */
__global__ void pixel_kernel(const float* __restrict__ mel,
                             const float* __restrict__ blood,
                             const float* __restrict__ shade,
                             const float* __restrict__ spec,
                             const float* __restrict__ params,
                             const float* __restrict__ P,
                             float* __restrict__ out, int imgStart)
{
  int local = blockIdx.x >> 4;                       /* 16 blocks / image */
  int n     = imgStart + local;
  int p     = ((blockIdx.x & 15) << 8) + threadIdx.x;
  const float* prm = params + (size_t)n*PARAM_STRIDE;
  size_t ip = (size_t)n*PIX + p;

  float mv = 2.0f/(1.0f+expf(-mel[ip]))   - 1.0f;
  float bv = 2.0f/(1.0f+expf(-blood[ip])) - 1.0f;
  float sh = expf(shade[ip]);

  float gx = (mv+1.0f)*0.5f*255.0f;
  float gy = (bv+1.0f)*0.5f*255.0f;
  float fx = floorf(gx), fy = floorf(gy);
  float wx = gx - fx,    wy = gy - fy;
  int X0 = min(max((int)fx,0),255), Y0 = min(max((int)fy,0),255);
  int X1 = min(X0+1,255),           Y1 = min(Y0+1,255);
  float w00=(1.f-wx)*(1.f-wy), w01=wx*(1.f-wy), w10=(1.f-wx)*wy, w11=wx*wy;

  const float* Pimg = P + (size_t)local*3*SKTEX;
  float wb[3];
  #pragma unroll
  for (int c=0;c<3;c++){
    const float* pl = Pimg + c*SKTEX;
    float d = pl[Y0*256+X0]*w00 + pl[Y0*256+X1]*w01
            + pl[Y1*256+X0]*w10 + pl[Y1*256+X1]*w11;
    float lc = prm[108+c];
    size_t cp = ((size_t)n*3 + c)*PIX + p;
    float sp = expf(spec[cp]) * lc;
    float r  = sh*d + sp;
    wb[c] = r * prm[111+c];
    out[SPEC_OFF + cp] = sp;
    out[RAW_OFF  + cp] = r;
  }
  #pragma unroll
  for (int o=0;o<3;o++){
    float v = prm[114+o*3+0]*wb[0] + prm[114+o*3+1]*wb[1] + prm[114+o*3+2]*wb[2];
    out[RGB_OFF + ((size_t)n*3 + o)*PIX + p] = fmaxf(v, 0.0f);
  }
  out[SHADE_OFF + ip] = sh;
  out[BLOOD_OFF + ip] = bv;
  out[MEL_OFF   + ip] = mv;
}

extern "C" void kernel_launch(void* const* d_in, const int* in_sizes, int n_in,
                              void* d_out, int out_size, void* d_ws, size_t ws_size,
                              hipStream_t stream)
{
  const float* lighting = (const float*)d_in[0];
  const float* mel      = (const float*)d_in[1];
  const float* blood    = (const float*)d_in[2];
  const float* shade    = (const float*)d_in[3];
  const float* spec     = (const float*)d_in[4];
  const float* illA     = (const float*)d_in[5];
  const float* illF     = (const float*)d_in[6];
  const float* pcaMeans = (const float*)d_in[7];
  const float* pcaComp  = (const float*)d_in[8];
  const float* skin     = (const float*)d_in[9];
  const float* tmatrix  = (const float*)d_in[10];
  float* out = (float*)d_out;

  const size_t paramBytes = (size_t)NIMG*PARAM_STRIDE*sizeof(float);
  const size_t imgBytes   = (size_t)3*SKTEX*sizeof(float);
  float* params = (float*)d_ws;
  float* P      = (float*)((char*)d_ws + paramBytes);

  int chunk = 1;
  if (ws_size > paramBytes){
    size_t c = (ws_size - paramBytes) / imgBytes;
    if (c < 1)    c = 1;
    if (c > NIMG) c = NIMG;
    chunk = (int)c;
  }

  setup_kernel<<<1, 128, 0, stream>>>(lighting, illA, illF, pcaMeans, pcaComp,
                                      tmatrix, params, out);
  for (int s = 0; s < NIMG; s += chunk){
    int c = (NIMG - s < chunk) ? (NIMG - s) : chunk;
    int groups = (c + IBATCH - 1) / IBATCH;
    dim3 g2(16, groups);
    gemm_kernel <<<g2,   256, 0, stream>>>(params, skin, P, s, c);
    pixel_kernel<<<c*16, 256, 0, stream>>>(mel, blood, shade, spec, params, P, out, s);
  }
}